// ReuseLinear_52166672777842
// MI455X (gfx1250) — compile-verified
//
#include <hip/hip_runtime.h>

typedef float v2f __attribute__((ext_vector_type(2)));
typedef float v4f __attribute__((ext_vector_type(4)));
typedef float v8f __attribute__((ext_vector_type(8)));
typedef unsigned int u32x4 __attribute__((ext_vector_type(4)));
typedef int i32x4 __attribute__((ext_vector_type(4)));
typedef int i32x8 __attribute__((ext_vector_type(8)));

#define B_  16
#define P_  128
#define D_  256
#define N_  256
#define BP_ (B_ * P_)       // 2048 (b,p) rows
#define TD  16              // d-channels per tile (WMMA M/N)
#define ROWPAD 260          // padded LDS row stride (floats) == TDM pad 4 DW / 256 DW
#define BP_PER_WG 8

// ---------------------------------------------------------------------------
// Kernel 1: scatter W through perm once:  Wp[d, perm[d,k]] = W[d,k]
// ---------------------------------------------------------------------------
__global__ __launch_bounds__(256) void permute_w_kernel(
    const float* __restrict__ W, const int* __restrict__ perm,
    float* __restrict__ Wp) {
  int d = blockIdx.x;
  int k = threadIdx.x;
  Wp[d * N_ + perm[d * N_ + k]] = W[d * N_ + k];
}

// ---------------------------------------------------------------------------
// TDM: async-load a contiguous 16x256 f32 tile (16KB) into LDS with
// hardware row padding (4 DWORDs every 256 DWORDs -> 260-float LDS rows).
// D# packing per CDNA5 ISA.
// This toolchain's builtin takes 6 args (extra i32x8 group, zero for 2-D).
// ---------------------------------------------------------------------------
__device__ __forceinline__ void tdm_load_tile16x256(const float* gsrc,
                                                    unsigned lds_byte_off) {
  unsigned long long ga = (unsigned long long)(uintptr_t)gsrc;

  u32x4 g0;
  g0[0] = 1u;                                   // count=1, user mode, no gather
  g0[1] = lds_byte_off;                         // lds_addr (bytes)
  g0[2] = (unsigned)(ga & 0xFFFFFFFFu);         // global_addr[31:0]
  g0[3] = (unsigned)((ga >> 32) & 0x01FFFFFFu)  // global_addr[56:32]
          | (2u << 30);                         // type=2 ("image")

  i32x8 g1;
  g1[0] = (2 << 16)      // data_size = 4 bytes
        | (1 << 20)      // pad_enable
        | (7 << 22)      // pad_interval: 256 DWORDs
        | (3 << 25);     // pad_amount:   4 DWORDs
  g1[1] = (N_ & 0xFFFF) << 16;   // tensor_dim0[15:0] = 256
  g1[2] = (TD << 16);            // tensor_dim0[31:16]=0 | tensor_dim1[15:0]=16
  g1[3] = (N_ << 16);            // tensor_dim1[31:16]=0 | tile_dim0 = 256
  g1[4] = TD;                    // tile_dim1 = 16, tile_dim2 = 0 (2-D)
  g1[5] = N_;                    // tensor_dim0_stride[31:0] = 256 (contiguous rows)
  g1[6] = 0;                     // stride hi / tensor_dim1_stride lo (unused, 2-D)
  g1[7] = 0;

  i32x4 gz4 = {0, 0, 0, 0};              // groups 2/3 unused for 2-D tiles
  i32x8 gz8 = {0, 0, 0, 0, 0, 0, 0, 0};  // extra group (clang-23 form), unused

  __builtin_amdgcn_tensor_load_to_lds(g0, g1, gz4, gz4, gz8, /*cpol=*/0);
}

// ---------------------------------------------------------------------------
// Kernel 2: streaming main pass.
//   y[bp, d0+i] = diag_i( X_tile(16xN) @ WpTile^T(Nx16) ) + bias
// 64 V_WMMA_F32_16X16X4_F32 per (bp, d-tile): 8 K-chunks per wave, D tiles
// dumped to LDS branch-free, diagonal gathered by 16 reduction threads.
// x tiles double-buffered through the Tensor Data Mover (TENSORcnt).
// ---------------------------------------------------------------------------
__global__ __launch_bounds__(256) void reuse_linear_main(
    const float* __restrict__ x, const float* __restrict__ Wp,
    const float* __restrict__ bias, float* __restrict__ out) {
  __shared__ float xs[2][TD * ROWPAD];   // double-buffered x tile (TDM dest)
  __shared__ float wps[TD * ROWPAD];     // Wp tile (TDM dest)
  __shared__ float dpart[8 * 32 * 8];    // full D tile per wave, lane-major
  __shared__ float yfin[TD];

  const int tid  = threadIdx.x;
  const int lane = tid & 31;
  const int wave = tid >> 5;
  const int d0   = blockIdx.x * TD;
  const int bp0  = blockIdx.y * BP_PER_WG;
  const float* xtile0 = x + ((size_t)bp0 * D_ + d0) * N_;

  // ---- prologue: wave 0 kicks off Wp tile + first x tile via TDM ----
  if (wave == 0) {
    tdm_load_tile16x256(Wp + (size_t)d0 * N_,
                        (unsigned)(uintptr_t)(&wps[0]));
    tdm_load_tile16x256(xtile0, (unsigned)(uintptr_t)(&xs[0][0]));
  }

  // WMMA f32 16x16x4 operand addressing (A row m == B column j == lane&15):
  //   lanes 0-15 hold K={0,1}; lanes 16-31 hold K={2,3}
  const int aoff = (lane & 15) * ROWPAD + (lane >> 4) * 2;

  for (int it = 0; it < BP_PER_WG; ++it) {
    // ---- issue next tile's DMA, retire current (double buffer) ----
    if (wave == 0) {
      if (it + 1 < BP_PER_WG) {
        tdm_load_tile16x256(xtile0 + (size_t)(it + 1) * (D_ * N_),
                            (unsigned)(uintptr_t)(&xs[(it + 1) & 1][0]));
        __builtin_amdgcn_s_wait_tensorcnt(1);   // current buffer complete
      } else {
        __builtin_amdgcn_s_wait_tensorcnt(0);
      }
    }
    __syncthreads();                            // buffer it&1 visible to all

    const float* xsb = &xs[it & 1][0];

    // ---- 8 K-chunks of 4 per wave: D += A(16x4) * B(4x16), fp32 WMMA ----
    v8f acc = {0.f, 0.f, 0.f, 0.f, 0.f, 0.f, 0.f, 0.f};
    #pragma unroll
    for (int c = 0; c < 8; ++c) {
      const int nc4 = (wave * 8 + c) * 4;
      v2f a, b;
      a.x = xsb[aoff + nc4];
      a.y = xsb[aoff + nc4 + 1];
      b.x = wps[aoff + nc4];
      b.y = wps[aoff + nc4 + 1];
      acc = __builtin_amdgcn_wmma_f32_16x16x4_f32(
          /*neg_a=*/false, a, /*neg_b=*/false, b,
          /*c_mod=*/(short)0, acc, /*reuse_a=*/false, /*reuse_b=*/false);
    }

    // ---- branch-free D dump: each lane stores its 8 D elements (2x b128) ----
    {
      float* dp = dpart + (wave * 32 + lane) * 8;
      v4f lo = {acc[0], acc[1], acc[2], acc[3]};
      v4f hi = {acc[4], acc[5], acc[6], acc[7]};
      *(v4f*)(dp)     = lo;
      *(v4f*)(dp + 4) = hi;
    }
    __syncthreads();

    // ---- gather diagonal + combine 8 wave partials + bias ----
    // D layout: vgpr r, lanes0-15 -> (M=r, N=lane); lanes16-31 -> (M=r+8, N=lane-16)
    if (tid < TD) {
      const int srclane = (tid < 8) ? tid : (tid + 16);
      const int r = tid & 7;
      float s = bias[d0 + tid];
      #pragma unroll
      for (int w = 0; w < 8; ++w)
        s += dpart[(w * 32 + srclane) * 8 + r];
      yfin[tid] = s;
    }
    __syncthreads();

    // ---- broadcast store: out[bp, d0+dd, :] = y (NT b128, contiguous groups) ----
    {
      const int bp = bp0 + it;
      const int dd = tid >> 4;
      const int cc = tid & 15;
      const float val = yfin[dd];
      v4f v = {val, val, val, val};
      float* ob = out + ((size_t)bp * D_ + (d0 + dd)) * N_;
      #pragma unroll
      for (int i = 0; i < 4; ++i)
        __builtin_nontemporal_store(v, (v4f*)(ob + cc * 4 + 64 * i));
    }
    // no trailing barrier needed: top-of-loop barrier orders buffer reuse
  }
}

// ---------------------------------------------------------------------------
extern "C" void kernel_launch(void* const* d_in, const int* in_sizes, int n_in,
                              void* d_out, int out_size, void* d_ws, size_t ws_size,
                              hipStream_t stream) {
  const float* x    = (const float*)d_in[0];   // [B,P,D,N] f32
  const float* W    = (const float*)d_in[1];   // [D,N]     f32
  const float* bias = (const float*)d_in[2];   // [D]       f32
  const int*   perm = (const int*)  d_in[3];   // [D,N]     i32
  float* out = (float*)d_out;                  // [B,P,D,N] f32
  float* Wp  = (float*)d_ws;                   // scratch: permuted W, 256KB

  permute_w_kernel<<<dim3(D_), dim3(N_), 0, stream>>>(W, perm, Wp);
  reuse_linear_main<<<dim3(D_ / TD, BP_ / BP_PER_WG), dim3(256), 0, stream>>>(
      x, Wp, bias, out);
}